// Hard_Extract_Pool_Cluster_64836826301211
// MI455X (gfx1250) — compile-verified
//
#include <hip/hip_runtime.h>

// Problem constants (match the reference; index=256 is fixed)
#define BATCH   32
#define SEQ     512
#define DMODEL  768
#define HEADS   12
#define CLUSTER 5
#define TOPN    250          // index-1-CLUSTER = 256-1-5
#define NEXTR   251          // TOPN + cls token
#define NTAIL   260          // 511 - 251 (rank 250 dropped)
#define GROUP   53           // ceil(260/5) after pad -> 265/5
#define NCH     24           // row chunks for colsum
#define ROWS_PER_CHUNK 256   // 6144 / 24
#define LDSROW  528          // 512 + 16 pad (kills bank conflicts, 64 banks)

typedef __attribute__((ext_vector_type(2))) float v2f;
typedef __attribute__((ext_vector_type(8))) float v8f;

// ---------------------------------------------------------------------------
// Kernel 1: column partial sums of atten via V_WMMA_F32_16X16X4_F32.
// atten viewed per batch as a (HEADS*SEQ = 6144) x 512 matrix.
// grid = (NCH, BATCH), block = 256 (8 waves).
// Each WG: 256 rows, all 512 columns. Stage 8 rows -> LDS (coalesced b128),
// then 8 waves x 4 strips x 2 WMMAs (A = ones => D[m][n] = sum_k B[k][n]).
// ---------------------------------------------------------------------------
__global__ __launch_bounds__(256) void colsum_wmma(
    const float* __restrict__ atten, float* __restrict__ partial)
{
    __shared__ float lds[8 * LDSROW];

    const int chunk = blockIdx.x;
    const int b     = blockIdx.y;
    const int tid   = threadIdx.x;
    const int wave  = tid >> 5;
    const int lane  = tid & 31;
    const int half  = lane >> 4;   // which 16-lane half
    const int c     = lane & 15;   // column within strip (N = lane%16 in B/D)

    const size_t base = ((size_t)b * HEADS * SEQ + (size_t)chunk * ROWS_PER_CHUNK) * SEQ;

    const v2f ones = {1.0f, 1.0f};
    v8f acc[4] = {v8f{}, v8f{}, v8f{}, v8f{}};

    for (int stage = 0; stage < ROWS_PER_CHUNK / 8; ++stage) {
        // stage 8 rows (16 KB) into LDS, fully coalesced float4 loads
        const float4* src = reinterpret_cast<const float4*>(atten + base + (size_t)stage * 8 * SEQ);
#pragma unroll
        for (int k = 0; k < 4; ++k) {
            int g   = tid + k * 256;     // 1024 float4 groups = 8 rows * 128
            int row = g >> 7;
            int cg  = g & 127;
            float4 v = src[g];
            *reinterpret_cast<float4*>(&lds[row * LDSROW + cg * 4]) = v;
        }
        __syncthreads();

#pragma unroll
        for (int q = 0; q < 4; ++q) {
            const int col = (wave + 8 * q) * 16 + c;
            // rows 0..3 : v0 <- rows {0,1}, v1 <- rows {2,3} (k-slot order is
            // irrelevant because A == ones)
            v2f b0 = { lds[(0 + half) * LDSROW + col], lds[(2 + half) * LDSROW + col] };
            // rows 4..7
            v2f b1 = { lds[(4 + half) * LDSROW + col], lds[(6 + half) * LDSROW + col] };
            acc[q] = __builtin_amdgcn_wmma_f32_16x16x4_f32(
                false, ones, false, b0, (short)0, acc[q], false, false);
            acc[q] = __builtin_amdgcn_wmma_f32_16x16x4_f32(
                false, ones, false, b1, (short)0, acc[q], false, false);
        }
        __syncthreads();
    }

    // D layout: lane L, VGPR v holds D[M][N=L%16]; all rows identical (A=ones)
    if (lane < 16) {
        float* p = partial + (size_t)(b * NCH + chunk) * SEQ;
#pragma unroll
        for (int q = 0; q < 4; ++q)
            p[(wave + 8 * q) * 16 + c] = acc[q][0];
    }
}

// ---------------------------------------------------------------------------
// Kernel 2: reduce chunk partials + subtract diagonal, scale by 1/HEADS.
// grid = BATCH, block = 512.
// ---------------------------------------------------------------------------
__global__ void finalize_scores(const float* __restrict__ partial,
                                const float* __restrict__ atten,
                                float* __restrict__ score)
{
    const int b = blockIdx.x;
    const int j = threadIdx.x;
    float s = 0.0f;
    for (int ch = 0; ch < NCH; ++ch)
        s += partial[(size_t)(b * NCH + ch) * SEQ + j];
    float diag = 0.0f;
    for (int h = 0; h < HEADS; ++h)
        diag += atten[((size_t)(b * HEADS + h) * SEQ + j) * SEQ + j];
    score[b * SEQ + j] = (s - diag) * (1.0f / (float)HEADS);
}

// ---------------------------------------------------------------------------
// Kernel 3: per-batch ranking (stable, ties -> lower index first, matching
// jax.lax.top_k) + index-sorted placement into gather lists.
//   rank <  250 -> top set   (position 1 + #{top idx < j}; slot 0 = token 0)
//   rank == 250 -> dropped
//   rank >  250 -> tail set  (position #{tail idx < j})
// grid = BATCH, block = 512.
// ---------------------------------------------------------------------------
__global__ void rank_select(const float* __restrict__ score,
                            int* __restrict__ gTop, int* __restrict__ gTail)
{
    __shared__ float s[SEQ];
    __shared__ int   rnk[SEQ];

    const int b = blockIdx.x;
    const int j = threadIdx.x;

    s[j] = score[b * SEQ + j];
    __syncthreads();

    int rank = 1 << 30;                 // token 0 excluded from scoring
    if (j >= 1) {
        rank = 0;
        const float v = s[j];
        for (int t = 1; t < SEQ; ++t) {
            float u = s[t];
            rank += (u > v) || (u == v && t < j);
        }
    }
    rnk[j] = rank;
    __syncthreads();

    if (j == 0) gTop[b * NEXTR] = 0;    // cls token, smallest index
    if (j >= 1 && rank != TOPN) {
        if (rank < TOPN) {
            int pos = 1;
            for (int t = 1; t < j; ++t) pos += (rnk[t] < TOPN);
            gTop[b * NEXTR + pos] = j;
        } else {
            int pos = 0;
            for (int t = 1; t < j; ++t) pos += (rnk[t] > TOPN);
            gTail[b * NTAIL + pos] = j;
        }
    }
}

// ---------------------------------------------------------------------------
// Kernel 4: gather extracted rows + cluster means -> output (B, 256, 768).
// grid = (256, BATCH), block = 192 (float4 per thread covers 768 cols).
// ---------------------------------------------------------------------------
__global__ __launch_bounds__(192) void gather_out(
    const float* __restrict__ x,
    const int* __restrict__ gTop, const int* __restrict__ gTail,
    float* __restrict__ out)
{
    const int p   = blockIdx.x;     // output row 0..255
    const int b   = blockIdx.y;
    const int col = threadIdx.x * 4;

    float4* o = reinterpret_cast<float4*>(out + ((size_t)(b * 256 + p)) * DMODEL + col);

    if (p < NEXTR) {
        int src = gTop[b * NEXTR + p];
        const float4 v = *reinterpret_cast<const float4*>(
            x + ((size_t)(b * SEQ + src)) * DMODEL + col);
        *o = v;
    } else {
        const int cl = p - NEXTR;   // 0..4
        float4 acc = {0.0f, 0.0f, 0.0f, 0.0f};
        for (int t = 0; t < GROUP; ++t) {
            int idx = cl * GROUP + t;
            if (idx < NTAIL) {
                int src = gTail[b * NTAIL + idx];
                const float4 v = *reinterpret_cast<const float4*>(
                    x + ((size_t)(b * SEQ + src)) * DMODEL + col);
                acc.x += v.x; acc.y += v.y; acc.z += v.z; acc.w += v.w;
            }
        }
        const float inv = 1.0f / (float)GROUP;   // pad rows count in divisor
        acc.x *= inv; acc.y *= inv; acc.z *= inv; acc.w *= inv;
        *o = acc;
    }
}

// ---------------------------------------------------------------------------
extern "C" void kernel_launch(void* const* d_in, const int* in_sizes, int n_in,
                              void* d_out, int out_size, void* d_ws, size_t ws_size,
                              hipStream_t stream)
{
    (void)in_sizes; (void)n_in; (void)out_size; (void)ws_size;

    const float* x     = (const float*)d_in[0];
    const float* atten = (const float*)d_in[1];
    // d_in[2] (index) is a compile-time constant 256 in the reference.
    float* out = (float*)d_out;

    char* ws = (char*)d_ws;
    float* partial = (float*)ws;                                     // 32*24*512 f32 = 1.5 MB
    float* score   = (float*)(ws + (size_t)BATCH * NCH * SEQ * 4);   // 32*512 f32
    int*   gTop    = (int*)(ws + (size_t)(BATCH * NCH * SEQ + BATCH * SEQ) * 4);
    int*   gTail   = gTop + BATCH * NEXTR;

    colsum_wmma   <<<dim3(NCH, BATCH), 256, 0, stream>>>(atten, partial);
    finalize_scores<<<BATCH, SEQ, 0, stream>>>(partial, atten, score);
    rank_select   <<<BATCH, SEQ, 0, stream>>>(score, gTop, gTail);
    gather_out    <<<dim3(256, BATCH), 192, 0, stream>>>(x, gTop, gTail, out);
}